// SpectralConvLaplace2D_85727547228520
// MI455X (gfx1250) — compile-verified
//
#include <hip/hip_runtime.h>

// SpectralConvLaplace2D on MI455X (gfx1250), fp32 complex, WMMA f32 16x16x4.
// B=8, CI=CO=32, M1=M2=16, H=W=128, TOTAL=288.

#define B_     8
#define CI_    32
#define CO_    32
#define H_     128
#define W_     128
#define TOTAL_ 288
#define PI_    3.14159265358979f
#define APAD_  18   // padded row stride for WMMA operand staging ([row][16] + 2)

typedef __attribute__((ext_vector_type(2))) float v2f;
typedef __attribute__((ext_vector_type(8))) float v8f;

// ---- workspace layout (float offsets) -------------------------------------
#define ALPHA_RE 0
#define ALPHA_IM 65536
#define U1_RE    131072
#define U1_IM    393216
#define U2_RE    655360
#define U2_IM    917504
#define S_RE     1179648
#define S_IM     1441792
#define OR1_RE   1703936
#define OR1_IM   1769472
#define OR2_RE   1835008
#define OR2_IM   1900544
// total floats: 1966080 (~7.5 MB)

// ===========================================================================
// Stage A: alpha[b,i,f,g] = sum_{y,x} x[y,x] e^{-2pi i f y/128} e^{-2pi i g x/128}
// One workgroup per (b,i) image; image staged through LDS in 32-row chunks so
// HBM sees x exactly once.
// ===========================================================================
__global__ void __launch_bounds__(256) alpha_kernel(const float* __restrict__ x,
                                                    float* __restrict__ ws) {
  __shared__ float cT[16 * 128], sT[16 * 128];
  __shared__ float simg[32 * 128];
  __shared__ float Tre[16 * 128], Tim[16 * 128];
  const int bi  = blockIdx.x;            // b*CI + i
  const int tid = threadIdx.x;
  const float* img = x + (size_t)bi * (H_ * W_);
  const float w0 = -2.0f * PI_ / 128.0f;

  for (int e = tid; e < 16 * 128; e += 256) {
    int f = e >> 7, y = e & 127;
    float ang = w0 * (float)(f * y);
    cT[e] = __cosf(ang);
    sT[e] = __sinf(ang);
  }

  // T[f][x] = sum_y img[y][x] * (c[f][y] + i s[f][y])   (x real)
  float accR[8], accI[8];
#pragma unroll
  for (int j = 0; j < 8; ++j) { accR[j] = 0.f; accI[j] = 0.f; }

  for (int yb = 0; yb < 4; ++yb) {
    __syncthreads();
    for (int e = tid; e < 32 * 128; e += 256)       // coalesced, read-once
      simg[e] = img[yb * (32 * 128) + e];
    __syncthreads();
#pragma unroll
    for (int j = 0; j < 8; ++j) {
      int e = tid + j * 256;
      int f = e >> 7, xc = e & 127;
      const float* crow = &cT[f * 128 + yb * 32];
      const float* srow = &sT[f * 128 + yb * 32];
      float re = accR[j], im = accI[j];
      for (int yy = 0; yy < 32; ++yy) {
        float v = simg[yy * 128 + xc];              // conflict-free (xc = lane)
        re += v * crow[yy];
        im += v * srow[yy];
      }
      accR[j] = re; accI[j] = im;
    }
  }
  __syncthreads();
#pragma unroll
  for (int j = 0; j < 8; ++j) {
    Tre[tid + j * 256] = accR[j];
    Tim[tid + j * 256] = accI[j];
  }
  __syncthreads();

  // alpha[f][g] = sum_x T[f][x] * (c[g][x] + i s[g][x])
  const int f = tid >> 4, g = tid & 15;
  float are = 0.f, aim = 0.f;
  for (int xc = 0; xc < 128; ++xc) {
    float tr = Tre[f * 128 + xc], ti = Tim[f * 128 + xc];
    float c = cT[g * 128 + xc], s = sT[g * 128 + xc];
    are += tr * c - ti * s;
    aim += tr * s + ti * c;
  }
  ws[ALPHA_RE + bi * 256 + tid] = are;
  ws[ALPHA_IM + bi * 256 + tid] = aim;
}

// ===========================================================================
// Stage B1: per (i,o): u1[f,p]=1/(lam1[f]-pole1[p]), u2[g,q]=1/(lam2[g]-pole2[q]),
//           S[f,g] = sum_{p,q} residue[p,q] u1[f,p] u2[g,q]
// ===========================================================================
__global__ void __launch_bounds__(256) precomp_kernel(const float* __restrict__ wr,
                                                      const float* __restrict__ wi,
                                                      float* __restrict__ ws) {
  __shared__ float u1re[256], u1im[256], u2re[256], u2im[256];
  __shared__ float Rre[256], Rim[256], T1re[256], T1im[256];
  const int io  = blockIdx.x;           // i*CO + o
  const int tid = threadIdx.x;
  const float* wrb = wr + (size_t)io * TOTAL_;
  const float* wib = wi + (size_t)io * TOTAL_;
  const int f = tid >> 4, p = tid & 15; // doubles as (g,q), (p,q), (f,g)

  // lam[f] = i * 2*pi*f*(H-1)/H  (purely imaginary; fftfreq(128, 1/127))
  const float lamc = 2.0f * PI_ * 127.0f / 128.0f;
  {
    float lam = lamc * (float)f;
    float dr = -wrb[p];
    float di = lam - wib[p];
    float inv = 1.0f / (dr * dr + di * di);
    u1re[tid] = dr * inv;
    u1im[tid] = -di * inv;
    float dr2 = -wrb[16 + p];
    float di2 = lam - wib[16 + p];
    float inv2 = 1.0f / (dr2 * dr2 + di2 * di2);
    u2re[tid] = dr2 * inv2;
    u2im[tid] = -di2 * inv2;
    Rre[tid] = wrb[32 + tid];
    Rim[tid] = wib[32 + tid];
  }
  __syncthreads();

  // T1[f][q] = sum_p u1[f][p] * R[p][q]     (tid = f*16 + q)
  {
    const int q = p;
    float re = 0.f, im = 0.f;
    for (int pp = 0; pp < 16; ++pp) {
      float ar = u1re[f * 16 + pp], ai = u1im[f * 16 + pp];
      float br = Rre[pp * 16 + q], bi = Rim[pp * 16 + q];
      re += ar * br - ai * bi;
      im += ar * bi + ai * br;
    }
    T1re[tid] = re; T1im[tid] = im;
  }
  __syncthreads();

  // S[f][g] = sum_q T1[f][q] * u2[g][q]     (tid = f*16 + g)
  {
    const int g = p;
    float re = 0.f, im = 0.f;
    for (int q = 0; q < 16; ++q) {
      float ar = T1re[f * 16 + q], ai = T1im[f * 16 + q];
      float br = u2re[g * 16 + q], bi = u2im[g * 16 + q];
      re += ar * br - ai * bi;
      im += ar * bi + ai * br;
    }
    ws[S_RE + io * 256 + tid] = re;
    ws[S_IM + io * 256 + tid] = im;
  }
  ws[U1_RE + io * 256 + tid] = u1re[tid];
  ws[U1_IM + io * 256 + tid] = u1im[tid];
  ws[U2_RE + io * 256 + tid] = u2re[tid];
  ws[U2_IM + io * 256 + tid] = u2im[tid];
}

// ===========================================================================
// Stage B2: per (b,o):
//   or1[f,g] = sum_i alpha[b,i,f,g] * S[f,g,i,o]
//   or2[p,q] = sum_i residue[i,o,p,q] * (u1^T . alpha[b,i] . u2)[p,q]
// ===========================================================================
__global__ void __launch_bounds__(256) or_kernel(const float* __restrict__ wr,
                                                 const float* __restrict__ wi,
                                                 float* __restrict__ ws) {
  __shared__ float sAre[256], sAim[256];
  __shared__ float su1re[256], su1im[256], su2re[256], su2im[256];
  __shared__ float sTre[256], sTim[256];
  const int bo  = blockIdx.x;           // b*CO + o
  const int b   = bo >> 5, o = bo & 31;
  const int tid = threadIdx.x;
  const int r = tid >> 4, c = tid & 15;

  float o1re = 0.f, o1im = 0.f, o2re = 0.f, o2im = 0.f;
  for (int i = 0; i < CI_; ++i) {
    const int io = i * CO_ + o;
    float are = ws[ALPHA_RE + (b * CI_ + i) * 256 + tid];
    float aim = ws[ALPHA_IM + (b * CI_ + i) * 256 + tid];
    sAre[tid] = are; sAim[tid] = aim;
    su1re[tid] = ws[U1_RE + io * 256 + tid];
    su1im[tid] = ws[U1_IM + io * 256 + tid];
    su2re[tid] = ws[U2_RE + io * 256 + tid];
    su2im[tid] = ws[U2_IM + io * 256 + tid];
    float sre = ws[S_RE + io * 256 + tid];
    float sim = ws[S_IM + io * 256 + tid];
    o1re += are * sre - aim * sim;       // or1 accumulation, elementwise in (f,g)
    o1im += are * sim + aim * sre;
    __syncthreads();

    // t1[p][g] = sum_f u1[f][p] * alpha[f][g]   (tid: p=r, g=c)
    float tre = 0.f, tim = 0.f;
    for (int ff = 0; ff < 16; ++ff) {
      float ur = su1re[ff * 16 + r], ui = su1im[ff * 16 + r];
      float ar = sAre[ff * 16 + c], ai = sAim[ff * 16 + c];
      tre += ur * ar - ui * ai;
      tim += ur * ai + ui * ar;
    }
    sTre[tid] = tre; sTim[tid] = tim;
    __syncthreads();

    // cpq = sum_g t1[p][g] * u2[g][q]           (tid: p=r, q=c)
    float cre = 0.f, cim = 0.f;
    for (int g = 0; g < 16; ++g) {
      float t1r = sTre[r * 16 + g], t1i = sTim[r * 16 + g];
      float ur = su2re[g * 16 + c], ui = su2im[g * 16 + c];
      cre += t1r * ur - t1i * ui;
      cim += t1r * ui + t1i * ur;
    }
    float rr = wr[(size_t)io * TOTAL_ + 32 + tid];
    float ri = wi[(size_t)io * TOTAL_ + 32 + tid];
    o2re += rr * cre - ri * cim;
    o2im += rr * cim + ri * cre;
    __syncthreads();
  }
  ws[OR1_RE + bo * 256 + tid] = o1re;
  ws[OR1_IM + bo * 256 + tid] = o1im;
  ws[OR2_RE + bo * 256 + tid] = o2re;
  ws[OR2_IM + bo * 256 + tid] = o2im;
}

// ===========================================================================
// Stage C: per (b,o):
//   out[z,x] = (1/HW) * Re( Ey^T.or1.Ex  +  sum_i e1[i,o]^T.or2.e2[i,o] )
// 33 terms, each = two real (128x16)x(16x128) matmuls on v_wmma_f32_16x16x4.
// Operand staging is TRANSPOSED + padded (row stride 18) so each lane's
// (K,K+1) pair is one aligned ds_load_b64, bank-conflict-free across the wave.
// B stores -Im so f32 WMMA needs no operand negation (ISA: NEG[1:0]==0 for F32).
// ===========================================================================
__global__ void __launch_bounds__(256) synth_kernel(const float* __restrict__ wr,
                                                    const float* __restrict__ wi,
                                                    const float* __restrict__ ws,
                                                    float* __restrict__ out) {
  __shared__ __align__(16) float sAre[128 * APAD_], sAim[128 * APAD_]; // [z][p]
  __shared__ float sEre[2048], sEim[2048];                             // [q][x]
  __shared__ __align__(16) float sBre[128 * APAD_], sBmi[128 * APAD_]; // [x][p]
  __shared__ float sO1re[256], sO1im[256], sO2re[256], sO2im[256];
  const int bo   = blockIdx.x;               // b*CO + o
  const int o    = bo & 31;
  const int tid  = threadIdx.x;
  const int wv   = tid >> 5;                 // wave id 0..7
  const int lane = tid & 31;

  sO1re[tid] = ws[OR1_RE + bo * 256 + tid];
  sO1im[tid] = ws[OR1_IM + bo * 256 + tid];
  sO2re[tid] = ws[OR2_RE + bo * 256 + tid];
  sO2im[tid] = ws[OR2_IM + bo * 256 + tid];

  v8f acc[8];
  {
    v8f zero = {0.f, 0.f, 0.f, 0.f, 0.f, 0.f, 0.f, 0.f};
#pragma unroll
    for (int t = 0; t < 8; ++t) acc[t] = zero;
  }

  const float tstep = 1.0f / 127.0f;         // ty = linspace(0,1,128)
  const float wf    = 2.0f * PI_ / 128.0f;   // ifft2 basis

  for (int term = 0; term <= CI_; ++term) {
    if (term == 0) {
      // Fourier term: A[f][z] = e^{+2pi i f z/128}, E[g][x] = e^{+2pi i g x/128}
      for (int e = tid; e < 2048; e += 256) {
        int rrow = e >> 7, idx = e & 127;
        float ang = wf * (float)(rrow * idx);
        float cv = __cosf(ang), sv = __sinf(ang);
        sAre[idx * APAD_ + rrow] = cv;       // transposed store
        sAim[idx * APAD_ + rrow] = sv;
        sEre[rrow * 128 + idx] = cv;
        sEim[rrow * 128 + idx] = sv;
      }
    } else {
      const int i = term - 1;
      const size_t wbase = (size_t)(i * CO_ + o) * TOTAL_;
      for (int e = tid; e < 2048; e += 256) {
        int rrow = e >> 7, idx = e & 127;
        float t = (float)idx * tstep;
        float p1r = wr[wbase + rrow], p1i = wi[wbase + rrow];
        float m1 = __expf(p1r * t);
        float a1 = p1i * t;
        sAre[idx * APAD_ + rrow] = m1 * __cosf(a1);  // e1[p][z], transposed
        sAim[idx * APAD_ + rrow] = m1 * __sinf(a1);
        float p2r = wr[wbase + 16 + rrow], p2i = wi[wbase + 16 + rrow];
        float m2 = __expf(p2r * t);
        float a2 = p2i * t;
        sEre[rrow * 128 + idx] = m2 * __cosf(a2);    // e2[q][x]
        sEim[rrow * 128 + idx] = m2 * __sinf(a2);
      }
    }
    __syncthreads();

    // B[p][x] = sum_q or[p][q] * E[q][x]  (complex); store transposed Re / -Im
    const float* oRe = (term == 0) ? sO1re : sO2re;
    const float* oIm = (term == 0) ? sO1im : sO2im;
    for (int e = tid; e < 2048; e += 256) {
      int rrow = e >> 7, xc = e & 127;
      float re = 0.f, im = 0.f;
      for (int q = 0; q < 16; ++q) {
        float cr = oRe[rrow * 16 + q], ci = oIm[rrow * 16 + q];
        float er = sEre[q * 128 + xc], ei = sEim[q * 128 + xc];
        re += cr * er - ci * ei;
        im += cr * ei + ci * er;
      }
      sBre[xc * APAD_ + rrow] = re;
      sBmi[xc * APAD_ + rrow] = -im;
    }
    __syncthreads();

    // acc += A_re^T.B_re + A_im^T.B_negim   (WMMA f32 16x16x4, K=16 in 4 steps)
    {
      const int m = lane & 15;
      const int z = wv * 16 + m;
      const int khalf = (lane >> 4) << 1;    // A/B ISA layout: lanes>=16 hold K+2,K+3
      v2f aRe[4], aIm[4];
#pragma unroll
      for (int kc = 0; kc < 4; ++kc) {
        int kb = kc * 4 + khalf;             // even -> 8B aligned pair
        aRe[kc] = *(const v2f*)&sAre[z * APAD_ + kb];
        aIm[kc] = *(const v2f*)&sAim[z * APAD_ + kb];
      }
#pragma unroll
      for (int xt = 0; xt < 8; ++xt) {
        int n = xt * 16 + (lane & 15);
#pragma unroll
        for (int kc = 0; kc < 4; ++kc) {
          int kb = kc * 4 + khalf;
          v2f bRe = *(const v2f*)&sBre[n * APAD_ + kb];
          v2f bMi = *(const v2f*)&sBmi[n * APAD_ + kb];
          acc[xt] = __builtin_amdgcn_wmma_f32_16x16x4_f32(
              false, aRe[kc], false, bRe, (short)0, acc[xt], false, false);
          acc[xt] = __builtin_amdgcn_wmma_f32_16x16x4_f32(
              false, aIm[kc], false, bMi, (short)0, acc[xt], false, false);
        }
      }
    }
    __syncthreads();
  }

  // store: D tile layout — VGPR r, lanes<16: M=r, lanes>=16: M=r+8; N=lane&15
  const float invHW = 1.0f / 16384.0f;
  const int zb = wv * 16 + ((lane >> 4) << 3);
  const int xl = lane & 15;
#pragma unroll
  for (int xt = 0; xt < 8; ++xt) {
    int xc = xt * 16 + xl;
#pragma unroll
    for (int r2 = 0; r2 < 8; ++r2) {
      out[((size_t)bo * 128 + zb + r2) * 128 + xc] = acc[xt][r2] * invHW;
    }
  }
}

// ===========================================================================
extern "C" void kernel_launch(void* const* d_in, const int* in_sizes, int n_in,
                              void* d_out, int out_size, void* d_ws, size_t ws_size,
                              hipStream_t stream) {
  (void)in_sizes; (void)n_in; (void)out_size; (void)ws_size;
  const float* x  = (const float*)d_in[0];
  const float* wr = (const float*)d_in[1];
  const float* wi = (const float*)d_in[2];
  float* out = (float*)d_out;
  float* ws  = (float*)d_ws;

  alpha_kernel  <<<B_ * CI_,  256, 0, stream>>>(x, ws);
  precomp_kernel<<<CI_ * CO_, 256, 0, stream>>>(wr, wi, ws);
  or_kernel     <<<B_ * CO_,  256, 0, stream>>>(wr, wi, ws);
  synth_kernel  <<<B_ * CO_,  256, 0, stream>>>(wr, wi, ws, out);
}